// MakeGrid_23063974379611
// MI455X (gfx1250) — compile-verified
//
#include <hip/hip_runtime.h>

// Voxel scatter-add for MI455X (gfx1250, wave32).
// grid[b, x, y, z, f] += features[p, f] for each in-box point p.
//
// Kernel 1: zero the 274.8 MB output with the CDNA5 async path:
//           LDS staging buffer + global_store_async_from_lds_b128 (ASYNCcnt).
// Kernel 2: one wave per point, lane == feature. Scalar (s_load) coords,
//           non-temporal feature loads, hardware no-return
//           global_atomic_add_f32 for the scatter.

constexpr float kMaxDist = 32.0f;
constexpr int   kBox     = 129;   // 2*32/0.5 + 1
constexpr int   kF       = 32;

// ---------------------------------------------------------------------------
// Zero-init via async LDS->global stores. Each lane owns a 16B zeroed LDS
// slot; every loop iteration fires one fire-and-forget b128 async store
// (512B per wave), pipelined on ASYNCcnt with no VGPR data dependency.
// ---------------------------------------------------------------------------
__global__ void __launch_bounds__(256) zero_grid_async(float* __restrict__ out,
                                                       int n4,   // # of float4
                                                       int n)    // # of floats
{
    __shared__ float4 zbuf[256];
    zbuf[threadIdx.x] = make_float4(0.f, 0.f, 0.f, 0.f);
    __syncthreads();

    // Wave-relative LDS byte offset = low 32 bits of the generic address
    // (flat LDS aperture).
    unsigned lds_off = (unsigned)(uintptr_t)(&zbuf[threadIdx.x]);

    long long stride = (long long)gridDim.x * 256;
    for (long long i = (long long)blockIdx.x * 256 + threadIdx.x; i < n4; i += stride) {
        const float4* gaddr = reinterpret_cast<const float4*>(out) + i;
        asm volatile("global_store_async_from_lds_b128 %0, %1, off"
                     :
                     : "v"(gaddr), "v"(lds_off)
                     : "memory");
    }
    // Drain this wave's outstanding async stores (S_ENDPGM would also wait,
    // but be explicit).
    asm volatile("s_wait_asynccnt 0x0" ::: "memory");

    // Scalar tail if out_size % 4 != 0 (not the case for 129^3*32, but safe).
    int tail = n - n4 * 4;
    if (blockIdx.x == 0 && (int)threadIdx.x < tail) {
        out[n4 * 4 + threadIdx.x] = 0.f;
    }
}

// ---------------------------------------------------------------------------
// Scatter: tid -> (point, feature). 32 lanes of a wave share one point:
// coords go through the scalar path (readfirstlane -> s_load), the feature
// load is one coalesced 128B NT transaction, and the 32 atomic adds hit one
// contiguous cacheline of the grid.
// ---------------------------------------------------------------------------
__global__ void __launch_bounds__(256) scatter_add(
    const float* __restrict__ coords,    // (N, 3)
    const float* __restrict__ features,  // (N, 32)
    float* __restrict__ grid,            // (129,129,129,32)
    int npoints)
{
    int tid = blockIdx.x * 256 + threadIdx.x;
    int p = tid >> 5;        // point index (uniform within a wave)
    int f = tid & 31;        // feature index == lane id
    if (p >= npoints) return;

    // p is wave-uniform: tell the compiler so the 12B coord triple becomes a
    // scalar load (s_load_b96, KMcnt) instead of 3 broadcast vector loads.
    int ps = __builtin_amdgcn_readfirstlane(p);
    const float* __restrict__ cbase = coords + (size_t)ps * 3;

    // Prefetch the feature stream ahead (global_prefetch_b8).
    int pp = p + 128;
    if (pp < npoints) __builtin_prefetch(&features[pp * kF + f], 0, 1);

    float cx = cbase[0];
    float cy = cbase[1];
    float cz = cbase[2];

    // jnp.round == round-half-to-even -> v_rndne_f32; /0.5 is exactly *2.0.
    float gx = __builtin_rintf((cx + kMaxDist) * 2.0f);
    float gy = __builtin_rintf((cy + kMaxDist) * 2.0f);
    float gz = __builtin_rintf((cz + kMaxDist) * 2.0f);

    int ix = (int)gx;
    int iy = (int)gy;
    int iz = (int)gz;

    // 0 <= i < 129 in one unsigned compare per axis.
    if ((unsigned)ix < (unsigned)kBox &&
        (unsigned)iy < (unsigned)kBox &&
        (unsigned)iz < (unsigned)kBox) {
        // Features are streamed exactly once: non-temporal load keeps the
        // 192MB L2 free for the grid the atomics are hammering.
        float v = __builtin_nontemporal_load(&features[p * kF + f]);

        int idx = ((ix * kBox + iy) * kBox + iz) * kF + f;   // < 68.7M, fits i32
        float* gptr = grid + idx;

        // Hardware no-return f32 atomic add (STOREcnt path).
        asm volatile("global_atomic_add_f32 %0, %1, off"
                     :
                     : "v"(gptr), "v"(v)
                     : "memory");
    }
}

// ---------------------------------------------------------------------------
extern "C" void kernel_launch(void* const* d_in, const int* in_sizes, int n_in,
                              void* d_out, int out_size, void* d_ws, size_t ws_size,
                              hipStream_t stream) {
    const float* coords   = (const float*)d_in[0];   // (1, N, 3) f32
    const float* features = (const float*)d_in[1];   // (1, N, 32) f32
    float* grid           = (float*)d_out;           // (1,129,129,129,32) f32

    int npoints = in_sizes[0] / 3;

    // 1) zero the grid (async LDS->global b128 stores, grid-stride)
    int n4 = out_size / 4;
    int zblocks = 16384;
    zero_grid_async<<<zblocks, 256, 0, stream>>>(grid, n4, out_size);

    // 2) scatter-add (1 thread per (point, feature))
    long long total = (long long)npoints * kF;
    int sblocks = (int)((total + 255) / 256);
    scatter_add<<<sblocks, 256, 0, stream>>>(coords, features, grid, npoints);
}